// GNN_node_60490319397093
// MI455X (gfx1250) — compile-verified
//
#include <hip/hip_runtime.h>

// Problem constants (match reference)
#define NN 100000
#define EE 600000
#define DIN 64
#define DE 8
#define DD 128
#define LL 5
#define BN_EPS 1e-5f

static_assert(NN % 16 == 0, "row tiles exact");
static_assert(EE % 8 == 0, "edge blocks exact");

typedef __attribute__((ext_vector_type(2))) float v2f;
typedef __attribute__((ext_vector_type(8))) float v8f;

// ---------------------------------------------------------------------------
// Zero-fill
// ---------------------------------------------------------------------------
__global__ __launch_bounds__(256) void zero_kernel(float* __restrict__ p, long n) {
    long i = (long)blockIdx.x * blockDim.x + threadIdx.x;
    if (i < n) p[i] = 0.0f;
}

// ---------------------------------------------------------------------------
// Edge kernel: one wave (32 lanes) per edge.
//   ee  = edge_attr[e] @ We_l + bee_l            (K = 8, D = 128)
//   msg = relu(h[src[e]] + ee)
//   atomicAdd(agg[dst[e]], msg)
// ---------------------------------------------------------------------------
__global__ __launch_bounds__(256) void edge_kernel(
    const float* __restrict__ h, const float* __restrict__ ea,
    const float* __restrict__ We_l, const float* __restrict__ bee_l,
    const int* __restrict__ src, const int* __restrict__ dst,
    float* __restrict__ agg)
{
    __shared__ float Ws[DE][DD];
    __shared__ float Bs[DD];
    for (int i = threadIdx.x; i < DE * DD; i += 256) Ws[i >> 7][i & 127] = We_l[i];
    if (threadIdx.x < DD) Bs[threadIdx.x] = bee_l[threadIdx.x];
    __syncthreads();

    const int e = blockIdx.x * 8 + (threadIdx.x >> 5);
    const int lane = threadIdx.x & 31;
    const int c0 = lane * 4;
    const int s = src[e];
    const int d = dst[e];

    float a0 = Bs[c0], a1 = Bs[c0 + 1], a2 = Bs[c0 + 2], a3 = Bs[c0 + 3];
#pragma unroll
    for (int k = 0; k < DE; ++k) {
        float av = ea[(long)e * DE + k];     // uniform per wave
        a0 += av * Ws[k][c0];
        a1 += av * Ws[k][c0 + 1];
        a2 += av * Ws[k][c0 + 2];
        a3 += av * Ws[k][c0 + 3];
    }
    const float4 hv = *(const float4*)(h + (long)s * DD + c0);
    a0 = fmaxf(a0 + hv.x, 0.0f);
    a1 = fmaxf(a1 + hv.y, 0.0f);
    a2 = fmaxf(a2 + hv.z, 0.0f);
    a3 = fmaxf(a3 + hv.w, 0.0f);
    float* ap = agg + (long)d * DD + c0;
    atomicAdd(ap + 0, a0);
    atomicAdd(ap + 1, a1);
    atomicAdd(ap + 2, a2);
    atomicAdd(ap + 3, a3);
}

// ---------------------------------------------------------------------------
// WMMA fp32 GEMM: Out[N,CN] = f(A)[N,K] @ B[K,CN] + bias
// Block = 256 threads = 8 waves; block owns 16 rows x 128 cols; each wave one
// 16x16 tile via V_WMMA_F32_16X16X4_F32. A tile staged in LDS with fused
// prologue:
//   MODE 0: a = A0                                   (node encoder)
//   MODE 1: a = (1+eps[l])*A0 + A1                   (GIN root update)
//   MODE 2: a = relu(A0*scale[k] + shift[k])         (hidden BN+ReLU)
// N % 16 == 0 and CN % 128 == 0, so no bounds guards anywhere.
// ---------------------------------------------------------------------------
template<int K, int CN, int MODE>
__global__ __launch_bounds__(256) void gemm_wmma_kernel(
    const float* __restrict__ A0, const float* __restrict__ A1,
    const float* __restrict__ A2,
    const float* __restrict__ epsArr, int l,
    const float* __restrict__ B, const float* __restrict__ bias,
    float* __restrict__ Out)
{
    static_assert(K % 4 == 0, "K multiple of 4");
    __shared__ float As[16][K + 1];   // +1 pad: LDS bank spread

    const int row0 = blockIdx.x * 16;
    const int wave = threadIdx.x >> 5;
    const int col0 = (blockIdx.y * 8 + wave) * 16;

    float alpha = 1.0f;
    if constexpr (MODE == 1) alpha = 1.0f + epsArr[l];

    // Stage/transform A tile (16 x K), coalesced along k
    for (int idx = threadIdx.x; idx < 16 * K; idx += 256) {
        const int r = idx / K;
        const int k = idx % K;
        const long off = (long)(row0 + r) * K + k;
        float v;
        if constexpr (MODE == 0) {
            v = A0[off];
        } else if constexpr (MODE == 1) {
            v = alpha * A0[off] + A1[off];
        } else {
            v = fmaxf(A0[off] * A1[k] + A2[k], 0.0f);
        }
        As[r][k] = v;
    }
    __syncthreads();

    const int lane  = threadIdx.x & 31;
    const int lr    = lane & 15;          // row of A frag / col of B frag
    const int khalf = (lane >> 4) * 2;    // lanes 16-31 carry K=+2,+3

    v8f acc = {};
    const float* Bp = B + col0 + lr;
#pragma unroll 4
    for (int k = 0; k < K; k += 4) {
        const int kk = k + khalf;
        v2f a;
        a.x = As[lr][kk];
        a.y = As[lr][kk + 1];
        v2f b;
        b.x = Bp[(long)kk * CN];
        b.y = Bp[(long)(kk + 1) * CN];
        acc = __builtin_amdgcn_wmma_f32_16x16x4_f32(
            false, a, false, b, (short)0, acc, false, false);
    }

    // Epilogue: lanes 0-15 hold rows M=0..7, lanes 16-31 rows M=8..15.
    const float bs = bias[col0 + lr];
    float* outp = Out + (long)(row0 + (lane >> 4) * 8) * CN + col0 + lr;
#pragma unroll
    for (int r = 0; r < 8; ++r) {
        outp[(long)r * CN] = acc[r] + bs;
    }
}

// ---------------------------------------------------------------------------
// BatchNorm stats stage 1: per-block partial column sums of X and X^2
// blockDim.x == C; sums layout: [0..C) = sum, [C..2C) = sumsq
// ---------------------------------------------------------------------------
__global__ void colstats_kernel(const float* __restrict__ X, float* __restrict__ sums,
                                int C, int Nrows, int rowsPerBlock)
{
    const int c = threadIdx.x;
    const int r0 = blockIdx.x * rowsPerBlock;
    const int r1 = min(r0 + rowsPerBlock, Nrows);
    float s = 0.0f, ss = 0.0f;
    for (int r = r0; r < r1; ++r) {
        float v = X[(long)r * C + c];
        s += v;
        ss += v * v;
    }
    atomicAdd(&sums[c], s);
    atomicAdd(&sums[C + c], ss);
}

// Fold BN into per-column scale/shift: y = x*scale + shift
__global__ void bn_finalize_kernel(const float* __restrict__ sums,
                                   const float* __restrict__ g, const float* __restrict__ b,
                                   float* __restrict__ scale, float* __restrict__ shift,
                                   int C, float invN)
{
    const int c = threadIdx.x + blockIdx.x * blockDim.x;
    if (c >= C) return;
    const float mu  = sums[c] * invN;
    const float var = sums[C + c] * invN - mu * mu;
    const float rs  = rsqrtf(var + BN_EPS);
    const float sc  = g[c] * rs;
    scale[c] = sc;
    shift[c] = b[c] - mu * sc;
}

// Outer BN apply (+optional relu): out = [relu](y*scale + shift), C = 128
__global__ __launch_bounds__(256) void bn_apply_kernel(
    const float* __restrict__ y, const float* __restrict__ scale,
    const float* __restrict__ shift, float* __restrict__ out, long n, int relu)
{
    const long i = (long)blockIdx.x * blockDim.x + threadIdx.x;
    if (i >= n) return;
    const int c = (int)(i & (DD - 1));
    float v = y[i] * scale[c] + shift[c];
    if (relu) v = fmaxf(v, 0.0f);
    out[i] = v;
}

// ---------------------------------------------------------------------------
// Launcher
// ---------------------------------------------------------------------------
extern "C" void kernel_launch(void* const* d_in, const int* in_sizes, int n_in,
                              void* d_out, int out_size, void* d_ws, size_t ws_size,
                              hipStream_t stream)
{
    const float* x    = (const float*)d_in[0];
    const int*   ei   = (const int*)  d_in[1];
    const float* ea   = (const float*)d_in[2];
    const float* Wn   = (const float*)d_in[3];
    const float* bn0  = (const float*)d_in[4];
    const float* We   = (const float*)d_in[5];
    const float* bee  = (const float*)d_in[6];
    const float* W1   = (const float*)d_in[7];
    const float* b1   = (const float*)d_in[8];
    const float* g1   = (const float*)d_in[9];
    const float* bt1  = (const float*)d_in[10];
    const float* W2   = (const float*)d_in[11];
    const float* b2   = (const float*)d_in[12];
    const float* eps  = (const float*)d_in[13];
    const float* gamma= (const float*)d_in[14];
    const float* beta = (const float*)d_in[15];

    const int* srcIdx = ei;
    const int* dstIdx = ei + EE;

    // Workspace layout (256B aligned)
    char* base = (char*)d_ws;
    size_t off = 0;
    auto take = [&](size_t bytes) {
        char* p = base + off;
        off = (off + bytes + 255) & ~(size_t)255;
        return p;
    };
    float* h      = (float*)take((size_t)NN * DD * 4);        // node features
    float* agg    = (float*)take((size_t)NN * DD * 4);        // scatter-add target
    float* t      = (float*)take((size_t)NN * 2 * DD * 4);    // MLP hidden pre-BN
    float* y      = (float*)take((size_t)NN * DD * 4);        // MLP out pre-BN
    float* stats  = (float*)take((size_t)768 * 4);            // [0,512)=stats1, [512,768)=stats2
    float* scale1 = (float*)take(2 * DD * 4);
    float* shift1 = (float*)take(2 * DD * 4);
    float* scale2 = (float*)take(DD * 4);
    float* shift2 = (float*)take(DD * 4);
    float* stats1 = stats;
    float* stats2 = stats + 512;
    (void)ws_size; (void)in_sizes; (void)n_in; (void)out_size;

    const float invN = 1.0f / (float)NN;
    const int ROWS_PER_STATS_BLOCK = 500;
    const int statsGrid = (NN + ROWS_PER_STATS_BLOCK - 1) / ROWS_PER_STATS_BLOCK;
    const dim3 gemmBlk(256);
    const dim3 gridEnc(NN / 16, 1);   // CN=128 -> 8 col-tiles = 1 y-slice
    const dim3 gridH(NN / 16, 2);     // CN=256 -> 16 col-tiles = 2 y-slices
    const dim3 gridO(NN / 16, 1);

    // Node encoder: h = x @ Wn + bn0   [N,64]x[64,128]
    gemm_wmma_kernel<DIN, DD, 0><<<gridEnc, gemmBlk, 0, stream>>>(
        x, nullptr, nullptr, nullptr, 0, Wn, bn0, h);

    for (int l = 0; l < LL; ++l) {
        // agg = 0 ; stats = 0
        {
            const long n = (long)NN * DD;
            zero_kernel<<<(unsigned)((n + 255) / 256), 256, 0, stream>>>(agg, n);
            zero_kernel<<<3, 256, 0, stream>>>(stats, 768);
        }

        // msg/scatter: agg[dst] += relu(h[src] + edge_attr@We[l] + bee[l])
        edge_kernel<<<EE / 8, 256, 0, stream>>>(
            h, ea, We + (size_t)l * DE * DD, bee + (size_t)l * DD,
            srcIdx, dstIdx, agg);

        // t = ((1+eps[l])*h + agg) @ W1[l] + b1[l]    [N,128]x[128,256]
        gemm_wmma_kernel<DD, 2 * DD, 1><<<gridH, gemmBlk, 0, stream>>>(
            h, agg, nullptr, eps, l,
            W1 + (size_t)l * DD * 2 * DD, b1 + (size_t)l * 2 * DD, t);

        // inner BN stats -> scale1/shift1
        colstats_kernel<<<statsGrid, 2 * DD, 0, stream>>>(t, stats1, 2 * DD, NN, ROWS_PER_STATS_BLOCK);
        bn_finalize_kernel<<<1, 2 * DD, 0, stream>>>(
            stats1, g1 + (size_t)l * 2 * DD, bt1 + (size_t)l * 2 * DD,
            scale1, shift1, 2 * DD, invN);

        // y = relu(bn(t)) @ W2[l] + b2[l]    [N,256]x[256,128]
        gemm_wmma_kernel<2 * DD, DD, 2><<<gridO, gemmBlk, 0, stream>>>(
            t, scale1, shift1, nullptr, 0,
            W2 + (size_t)l * 2 * DD * DD, b2 + (size_t)l * DD, y);

        // outer BN stats -> scale2/shift2
        colstats_kernel<<<statsGrid, DD, 0, stream>>>(y, stats2, DD, NN, ROWS_PER_STATS_BLOCK);
        bn_finalize_kernel<<<1, DD, 0, stream>>>(
            stats2, gamma + (size_t)l * DD, beta + (size_t)l * DD,
            scale2, shift2, DD, invN);

        // h = [relu](bn(y));  last layer writes d_out (no relu)
        float* dstp = (l == LL - 1) ? (float*)d_out : h;
        const long n = (long)NN * DD;
        bn_apply_kernel<<<(unsigned)((n + 255) / 256), 256, 0, stream>>>(
            y, scale2, shift2, dstp, n, (l < LL - 1) ? 1 : 0);
    }
}